// SpectralModule_12094627906090
// MI455X (gfx1250) — compile-verified
//
#include <hip/hip_runtime.h>
#include <hip/hip_bf16.h>

typedef __attribute__((ext_vector_type(2))) float v2f;
typedef __attribute__((ext_vector_type(8))) float v8f;

#define PATCH 13
#define EMBED 512
#define NTILES (EMBED / 16)

// cos/sin(2*pi*m/13), m = 0..12.  rfft uses exp(-i*2*pi*k*t/13); the sign of
// the imaginary part is irrelevant for |.|, and (k*t) mod 13 indexes this table.
__device__ __constant__ float DFT_COS[13] = {
     1.0f,
     0.8854560256532099f,  0.5680647467311558f,  0.12053668025532305f,
    -0.3546048870425356f, -0.7485107481711011f, -0.9709418174260520f,
    -0.9709418174260520f, -0.7485107481711011f, -0.3546048870425356f,
     0.12053668025532305f, 0.5680647467311558f,  0.8854560256532099f
};
__device__ __constant__ float DFT_SIN[13] = {
     0.0f,
     0.4647231720437686f,  0.8229838658936564f,  0.9927088740980539f,
     0.9350162426854148f,  0.6631226582407952f,  0.23931566428755777f,
    -0.23931566428755777f, -0.6631226582407952f, -0.9350162426854148f,
    -0.9927088740980539f, -0.8229838658936564f, -0.4647231720437686f
};

__global__ __launch_bounds__(256) void spectral_wmma_kernel(
    const float* __restrict__ x,     // [rows, 13]
    const float* __restrict__ W,     // [7, 512] row-major (in, out)
    const float* __restrict__ bvec,  // [512]
    float* __restrict__ out,         // [rows, 512]
    int rows)
{
    const int lane = threadIdx.x & 31;
    const int wave = blockIdx.x * (blockDim.x >> 5) + (threadIdx.x >> 5);
    const int row_base = wave * 16;
    if (row_base >= rows) return;               // wave-uniform: EXEC stays all-ones

    // ---- 13-point real DFT -> 7 magnitude bins, for row (lane & 15) ----
    const int r = row_base + (lane & 15);       // both half-waves compute same rows
    const float* xr = x + (size_t)r * PATCH;
    float xs[PATCH];
    #pragma unroll
    for (int t = 0; t < PATCH; ++t) xs[t] = xr[t];

    float mag[7];
    #pragma unroll
    for (int k = 0; k < 7; ++k) {
        float re = 0.0f, im = 0.0f;
        #pragma unroll
        for (int t = 0; t < PATCH; ++t) {
            const int m = (k * t) % PATCH;      // compile-time after unroll
            re = __builtin_fmaf(xs[t], DFT_COS[m], re);
            im = __builtin_fmaf(xs[t], DFT_SIN[m], im);
        }
        mag[k] = sqrtf(__builtin_fmaf(re, re, im * im)) * (1.0f / 13.0f);
    }

    // ---- A fragments for V_WMMA_F32_16X16X4_F32 (16x4 f32 = 2 VGPRs) ----
    // Lanes 0-15: M=0..15, VGPR0=K, VGPR1=K+1 ; lanes 16-31: K+2, K+3.
    const bool hi = (lane >= 16);
    v2f a0, a1;
    a0.x = hi ? mag[2] : mag[0];
    a0.y = hi ? mag[3] : mag[1];
    a1.x = hi ? mag[6] : mag[4];
    a1.y = hi ? 0.0f   : mag[5];                // K=7 zero pad

    const int col0 = lane & 15;
    const size_t orow = (size_t)(row_base + (hi ? 8 : 0)) * EMBED;

    for (int jt = 0; jt < NTILES; ++jt) {
        const int col = jt * 16 + col0;

        // B fragments (4x16 f32 = 2 VGPRs): row k striped across lanes.
        // All four loads are unconditionally in-bounds (rows 0..6 of W exist
        // for every col); the K=7 pad is a post-load select, NOT a predicated
        // load -- avoids s_and_saveexec/branch in the hot loop.
        v2f b0, b1;
        b0.x = W[(hi ? 2 : 0) * EMBED + col];
        b0.y = W[(hi ? 3 : 1) * EMBED + col];
        b1.x = W[(hi ? 6 : 4) * EMBED + col];
        const float w5 = W[5 * EMBED + col];    // in-bounds for all lanes
        b1.y = hi ? 0.0f : w5;                  // K=7 zero pad

        // Seed the accumulator with the bias (uniform over the 8 C rows this
        // lane holds, since bias depends only on N=col).  Removes 8 post-WMMA
        // adds from the store critical path.
        const float bias = bvec[col];
        v8f c;
        #pragma unroll
        for (int v = 0; v < 8; ++v) c[v] = bias;

        c = __builtin_amdgcn_wmma_f32_16x16x4_f32(false, a0, false, b0,
                                                  (short)0, c, false, false);
        c = __builtin_amdgcn_wmma_f32_16x16x4_f32(false, a1, false, b1,
                                                  (short)0, c, false, false);

        float* o = out + orow + col;
        // C/D layout: VGPR v holds (M = v [+8 for hi lanes], N = lane&15).
        // 512 MB streaming output -> non-temporal stores, don't pollute L2.
        #pragma unroll
        for (int v = 0; v < 8; ++v) {
            __builtin_nontemporal_store(c[v], o + (size_t)v * EMBED);
        }
    }
}

extern "C" void kernel_launch(void* const* d_in, const int* in_sizes, int n_in,
                              void* d_out, int out_size, void* d_ws, size_t ws_size,
                              hipStream_t stream)
{
    const float* x    = (const float*)d_in[0];   // [32,64,128,13]
    const float* W    = (const float*)d_in[1];   // [7,512]
    const float* bias = (const float*)d_in[2];   // [512]
    float* out        = (float*)d_out;           // [32,64,128,512]

    const int rows  = in_sizes[0] / PATCH;       // 262144
    const int waves = rows / 16;                 // 16384
    const int wavesPerBlock = 8;                 // 256 threads, wave32
    const int blocks = (waves + wavesPerBlock - 1) / wavesPerBlock;

    spectral_wmma_kernel<<<blocks, 256, 0, stream>>>(x, W, bias, out, rows);
}